// LinformerSpatialAttentionModule_70033736728661
// MI455X (gfx1250) — compile-verified
//
#include <hip/hip_runtime.h>
#include <math.h>

typedef __attribute__((ext_vector_type(2))) float v2f;
typedef __attribute__((ext_vector_type(8))) float v8f;

#define HEADS 4
#define KB 8
#define B_ 8
#define N_ 10000
#define H_ 512
#define HD 128
#define NCHUNK 40
#define ROWSPC 250   // N_/NCHUNK

// workspace layout (float offsets)
#define WS_PARTIAL 0
#define WS_XSUM   (WS_PARTIAL + B_*NCHUNK*H_)   // 163840
#define WS_WQ     (WS_XSUM + B_*H_)             // wq_eff [b][h][512]
#define WS_WO     (WS_WQ + B_*HEADS*H_)         // wo_pair [b][c][h] (head-pairs contiguous)
#define WS_QOFF   (WS_WO + B_*H_*HEADS)         // [b][h]
#define WS_EP     (WS_QOFF + B_*HEADS)          // Esum/sqrt(128) [h][kk]
#define WS_FS     (WS_EP + HEADS*KB)            // Fsum [h][kk]
#define WS_BIAS   (WS_FS + HEADS*KB)            // diag(gbu@gbv) [h][kk]
#define WS_ZERO   (WS_BIAS + HEADS*KB)          // 512 zeros (pad rows for WMMA A)

// ---------- K1: partial column sums of x ----------
__global__ void __launch_bounds__(512) k_xpartial(const float* __restrict__ x,
                                                  float* __restrict__ ws) {
    int ch = blockIdx.x, b = blockIdx.y, c = threadIdx.x;
    const float* p = x + ((size_t)(b * N_ + ch * ROWSPC)) * H_ + c;
    float s = 0.f;
    for (int r = 0; r < ROWSPC; ++r) s += p[(size_t)r * H_];
    ws[WS_PARTIAL + (b * NCHUNK + ch) * H_ + c] = s;
}

// ---------- K2: finalize xsum, zero the WMMA pad region ----------
__global__ void __launch_bounds__(512) k_xsum(float* __restrict__ ws) {
    int b = blockIdx.x, c = threadIdx.x;
    float s = 0.f;
    for (int ch = 0; ch < NCHUNK; ++ch) s += ws[WS_PARTIAL + (b * NCHUNK + ch) * H_ + c];
    ws[WS_XSUM + b * H_ + c] = s;
    if (b == 0) ws[WS_ZERO + c] = 0.f;
}

// ---------- K3: Esum/Fsum over N, graph-bias diagonal, reg scalar ----------
__global__ void __launch_bounds__(256) k_small(const float* __restrict__ E,
                                               const float* __restrict__ Fp,
                                               const float* __restrict__ gbu,
                                               const float* __restrict__ gbv,
                                               const float* __restrict__ lr,
                                               float* __restrict__ ws,
                                               float* __restrict__ out_reg) {
    __shared__ float red[256];
    int t = threadIdx.x;
    // 64 outputs (32 Esum + 32 Fsum), 4 threads each
    int oid = t >> 2, sub = t & 3;
    int o = oid & 31, h = o >> 3, kk = o & 7;
    const float* src = (oid < 32) ? E : Fp;
    float s = 0.f;
    for (int i = sub; i < N_; i += 4) s += src[(h * N_ + i) * KB + kk];
    red[t] = s;
    __syncthreads();
    if (sub == 0) {
        float tot = red[t] + red[t + 1] + red[t + 2] + red[t + 3];
        if (oid < 32) ws[WS_EP + o] = tot * 0.088388347648318440f; // 1/sqrt(128)
        else          ws[WS_FS + o] = tot;
    }
    __syncthreads();
    // graph_bias[h][a][bb], 256 elements, one per thread
    int h2 = t >> 6, a = (t >> 3) & 7, bb = t & 7;
    float g = 0.f;
    for (int j = 0; j < 8; ++j) g += gbu[(h2 * 8 + a) * 8 + j] * gbv[(h2 * 8 + j) * 8 + bb];
    if (a == bb) ws[WS_BIAS + h2 * 8 + a] = g;
    red[t] = fabsf(g);
    __syncthreads();
    for (int st = 128; st > 0; st >>= 1) {
        if (t < st) red[t] += red[t + st];
        __syncthreads();
    }
    // mean(|attn|) == 1/8 exactly (softmax rows sum to 1)
    if (t == 0) out_reg[0] = __expf(lr[0]) * (0.125f + (red[0] / 256.f) * 0.1f);
}

// ---------- K4: per (b,h) effective vectors ----------
__global__ void __launch_bounds__(256) k_prep(const float* __restrict__ Wq,
                                              const float* __restrict__ bq,
                                              const float* __restrict__ Wk,
                                              const float* __restrict__ bk,
                                              const float* __restrict__ Wv,
                                              const float* __restrict__ bv,
                                              const float* __restrict__ Wo,
                                              float* __restrict__ ws) {
    int b = blockIdx.x >> 2, h = blockIdx.x & 3, t = threadIdx.x;
    __shared__ float ks[HD], vs[HD];
    const float* xs = ws + WS_XSUM + b * H_;
    if (t < 128) {
        int row = h * HD + t;
        const float* w = Wk + (size_t)row * H_;
        float s = 0.f;
        for (int c = 0; c < H_; ++c) s += xs[c] * w[c];
        ks[t] = s + (float)N_ * bk[row];
    } else {
        int d = t - 128, row = h * HD + d;
        const float* w = Wv + (size_t)row * H_;
        float s = 0.f;
        for (int c = 0; c < H_; ++c) s += xs[c] * w[c];
        vs[d] = s + (float)N_ * bv[row];
    }
    __syncthreads();
    for (int c = t; c < H_; c += 256) {
        float sq = 0.f, so = 0.f;
        for (int d = 0; d < HD; ++d) {
            sq += ks[d] * Wq[(size_t)(h * HD + d) * H_ + c];
            so += vs[d] * Wo[(size_t)c * H_ + h * HD + d];
        }
        ws[WS_WQ + (b * HEADS + h) * H_ + c] = sq;
        ws[WS_WO + (b * H_ + c) * HEADS + h] = so;
    }
    if (t == 0) {
        float qo = 0.f;
        for (int d = 0; d < HD; ++d) qo += ks[d] * bq[h * HD + d];
        ws[WS_QOFF + b * HEADS + h] = qo;
    }
}

// ---------- K5: fused main pass, one 16x512 tile per wave32, WMMA f32 16x16x4 ----------
__global__ void __launch_bounds__(256) k_main(const float* __restrict__ x,
                                              const float* __restrict__ bo,
                                              const float* __restrict__ ws,
                                              float* __restrict__ out) {
    const int lane = threadIdx.x & 31;
    const int wave = (blockIdx.x << 3) + (threadIdx.x >> 5); // 0..4999
    const int b = wave / 625;
    const int n0 = (wave - b * 625) << 4;
    const int half = lane >> 4;   // 0: K pair {0,1}, 1: K pair {2,3}
    const int lmod = lane & 15;

    // ---- GEMM1: sT[16x16] = Wq_pad[16xK] * X^T[Kx16], K=512, chained accumulation
    // A(16x4) lane layout: VGPR0=A[M=lmod, k0+2*half], VGPR1=A[M=lmod, k0+2*half+1]
    // B(4x16) lane layout: VGPR0=B[k0+2*half, N=lmod],  VGPR1=B[k0+2*half+1, N=lmod]
    const float* xb = x + ((size_t)(b * N_ + n0 + lmod)) * H_ + 2 * half;
    const float* wqb = (lmod < HEADS)
        ? (ws + WS_WQ + (b * HEADS + lmod) * H_ + 2 * half)
        : (ws + WS_ZERO + 2 * half); // zeroed pad rows keep EXEC uniform
    v8f acc = {};
#pragma unroll 4
    for (int k0 = 0; k0 < H_; k0 += 4) {
        v2f a  = *(const v2f*)(wqb + k0);
        v2f bx = *(const v2f*)(xb + k0);
        acc = __builtin_amdgcn_wmma_f32_16x16x4_f32(false, a, false, bx,
                                                    (short)0, acc, false, false);
    }
    // lanes 0..15: acc[h] = sT[h, row=lane] for h=0..3 (rows 4..15 are zero pad)

    // ---- per-row softmax over kk=8, collapse to scalar weight per head
    const float* ep = ws + WS_EP;
    const float* fs = ws + WS_FS;
    const float* bi = ws + WS_BIAS;
    const float* qo = ws + WS_QOFF + b * HEADS;
    float wgt[4];
#pragma unroll
    for (int h = 0; h < 4; ++h) {
        float tv = acc[h] + qo[h];
        float sc[8];
        float m = -3.4e38f;
#pragma unroll
        for (int kk = 0; kk < 8; ++kk) {
            sc[kk] = ep[h * 8 + kk] * tv + bi[h * 8 + kk];
            m = fmaxf(m, sc[kk]);
        }
        float sp = 0.f, spf = 0.f;
#pragma unroll
        for (int kk = 0; kk < 8; ++kk) {
            float p = __expf(sc[kk] - m);
            sp += p;
            spf += p * fs[h * 8 + kk];
        }
        wgt[h] = spf / sp;
    }
    // rearrange w[row,h] into WMMA A(16x4) layout via lane permute
    int pidx = lmod << 2;
    float t0 = __int_as_float(__builtin_amdgcn_ds_bpermute(pidx, __float_as_int(wgt[0])));
    float t1 = __int_as_float(__builtin_amdgcn_ds_bpermute(pidx, __float_as_int(wgt[1])));
    float t2 = __int_as_float(__builtin_amdgcn_ds_bpermute(pidx, __float_as_int(wgt[2])));
    float t3 = __int_as_float(__builtin_amdgcn_ds_bpermute(pidx, __float_as_int(wgt[3])));
    v2f aw;
    aw.x = half ? t2 : t0;
    aw.y = half ? t3 : t1;

    // ---- GEMM2: Out[16x512] = w[16x4] * Wo_eff[4x512] + bo, K=4 exactly one WMMA per 16 cols
    const float* wob = ws + WS_WO + (b * H_ + lmod) * HEADS + 2 * half;
    float* outb = out + ((size_t)(b * N_ + n0 + 8 * half)) * H_ + lmod;
    for (int cb = 0; cb < H_; cb += 16) {
        v2f wo2 = *(const v2f*)(wob + cb * HEADS);
        float bv = bo[cb + lmod];
        v8f cm = {bv, bv, bv, bv, bv, bv, bv, bv};
        v8f d = __builtin_amdgcn_wmma_f32_16x16x4_f32(false, aw, false, wo2,
                                                      (short)0, cm, false, false);
#pragma unroll
        for (int g = 0; g < 8; ++g) outb[(size_t)g * H_ + cb] = d[g];
    }
}

extern "C" void kernel_launch(void* const* d_in, const int* in_sizes, int n_in,
                              void* d_out, int out_size, void* d_ws, size_t ws_size,
                              hipStream_t stream) {
    const float* x   = (const float*)d_in[0];
    const float* Wq  = (const float*)d_in[1];
    const float* bq  = (const float*)d_in[2];
    const float* Wk  = (const float*)d_in[3];
    const float* bk  = (const float*)d_in[4];
    const float* Wv  = (const float*)d_in[5];
    const float* bv  = (const float*)d_in[6];
    const float* E   = (const float*)d_in[7];
    const float* Fp  = (const float*)d_in[8];
    const float* gbu = (const float*)d_in[9];
    const float* gbv = (const float*)d_in[10];
    const float* Wo  = (const float*)d_in[11];
    const float* bo  = (const float*)d_in[12];
    const float* lr  = (const float*)d_in[13];
    float* out = (float*)d_out;
    float* ws  = (float*)d_ws;
    float* out_reg = out + (size_t)B_ * N_ * H_; // reg scalar after [B,N,H] output

    k_xpartial<<<dim3(NCHUNK, B_), 512, 0, stream>>>(x, ws);
    k_xsum<<<B_, 512, 0, stream>>>(ws);
    k_small<<<1, 256, 0, stream>>>(E, Fp, gbu, gbv, lr, ws, out_reg);
    k_prep<<<32, 256, 0, stream>>>(Wq, bq, Wk, bk, Wv, bv, Wo, ws);
    k_main<<<625, 256, 0, stream>>>(x, bo, ws, out);
}